// AttentionModel_75892072120845
// MI455X (gfx1250) — compile-verified
//
#include <hip/hip_runtime.h>
#include <hip/hip_bf16.h>
#include <math.h>

// ---------------------------------------------------------------------------
// AttentionModel greedy decode for MI455X (gfx1250, wave32).
//
// Precompute (TDM + WMMA f32 GEMM):
//   Wbig[256,1024] = [ W_node[:,0:256] | W_node[:,256:512] |
//                      W_node[:,512:768] @ W_out^T | W_step ]
//   [K | V | LK2 | E2][B,G,256] = embeddings[B*G,256] @ Wbig
//     - A/B tiles staged global->LDS by TENSOR_LOAD_TO_LDS (double-buffered,
//       descriptor pad feature produces the padded LDS layout directly)
//     - math via v_wmma_f32_16x16x4_f32
//   graph_embed -> fixed_ctx, q0base = fixed_ctx + graph_embed@W_step
//
// Decode: one persistent workgroup per batch element runs all n_steps with
// only workgroup barriers (state is per-batch). All step tensors stay in the
// 192MB L2 (K+V+LK2+E2 = 168MB).
// ---------------------------------------------------------------------------

#define B_  64
#define G_  640
#define D_  256
#define H_  8
#define DH_ 32

static constexpr float NEG_INF_F   = -1e9f;
static constexpr float TANH_CLIP_F = 10.0f;
static constexpr float INV_SQRT_DH = 0.17677669529663687f; // 1/sqrt(32)
static constexpr float INV_SQRT_D  = 0.0625f;              // 1/sqrt(256)

typedef float v2f __attribute__((ext_vector_type(2)));
typedef float v8f __attribute__((ext_vector_type(8)));
typedef unsigned int v4u __attribute__((ext_vector_type(4)));
typedef int v8i __attribute__((ext_vector_type(8)));
typedef int v4i __attribute__((ext_vector_type(4)));

// ---------------------------------------------------------------------------
// Weight prep: Wbig[d][0:256]=W_node K cols, [256:512]=V cols,
// [512:768]=W_node_lk @ W_out^T, [768:1024]=W_step.
// ---------------------------------------------------------------------------
__global__ __launch_bounds__(256) void wprep_kernel(
    const float* __restrict__ W_node, const float* __restrict__ W_step,
    const float* __restrict__ W_out, float* __restrict__ Wbig) {
  __shared__ float wrow[D_];
  const int d = blockIdx.x, tid = threadIdx.x;
  wrow[tid] = W_node[(size_t)d * 768 + 512 + tid];
  Wbig[(size_t)d * 1024 + tid]       = W_node[(size_t)d * 768 + tid];
  Wbig[(size_t)d * 1024 + 256 + tid] = W_node[(size_t)d * 768 + 256 + tid];
  Wbig[(size_t)d * 1024 + 768 + tid] = W_step[(size_t)d * 256 + tid];
  __syncthreads();
  float acc = 0.f;
  #pragma unroll 4
  for (int k = 0; k < D_; ++k) acc += wrow[k] * W_out[(size_t)tid * 256 + k];
  Wbig[(size_t)d * 1024 + 512 + tid] = acc;  // Wc[d][e]
}

// ---------------------------------------------------------------------------
// graph_embed -> fixed_ctx, q0base. grid=B, block=256 (d)
// ---------------------------------------------------------------------------
__global__ __launch_bounds__(256) void ctx_kernel(
    const float* __restrict__ emb, const float* __restrict__ W_fixed,
    const float* __restrict__ W_step, float* __restrict__ fixed_ctx,
    float* __restrict__ q0base) {
  __shared__ float ge[D_];
  const int b = blockIdx.x, d = threadIdx.x;
  float s = 0.f;
  for (int g = 0; g < G_; ++g) s += emb[((size_t)b * G_ + g) * D_ + d];
  ge[d] = s * (1.0f / G_);
  __syncthreads();
  float fc = 0.f, qs = 0.f;
  #pragma unroll 4
  for (int k = 0; k < D_; ++k) {
    const float gk = ge[k];
    fc += gk * W_fixed[(size_t)k * D_ + d];
    qs += gk * W_step[(size_t)k * D_ + d];
  }
  fixed_ctx[(size_t)b * D_ + d] = fc;
  q0base[(size_t)b * D_ + d]    = fc + qs;
}

// ---------------------------------------------------------------------------
// TDM helper: issue a 2-D tensor tile load global->LDS.
// D# packing per CDNA5 ISA 8.3-8.5: group0 = {count/type, lds_addr,
// global_addr}, group1 = {data_size/pad, tensor dims, tile dims, stride}.
// data_size = 4 bytes (code 2). Pad codes: interval 2^(c+1) dwords, amount
// (c+1) dwords -> reproduces padded LDS rows for bank-conflict avoidance.
// This toolchain's builtin takes 6 args (g0, g1, g2, g3, g4, cpol); the
// trailing descriptor group is unused for plain 2-D tiles -> zeros.
// ---------------------------------------------------------------------------
__device__ __forceinline__ unsigned lds_addr_of(const void* p) {
  // Generic (flat) pointers to LDS carry the LDS byte offset in the low 32b.
  return (unsigned)(size_t)p;
}

__device__ __forceinline__ void tdm_load_2d(
    const void* gptr, unsigned lds,
    unsigned tensor_d0, unsigned tensor_d1,
    unsigned tile_d0, unsigned tile_d1, unsigned stride0,
    unsigned pad_interval_code, unsigned pad_amount_code) {
  const unsigned long long ga = (unsigned long long)(size_t)gptr;
  v4u g0;
  g0[0] = 1u;                                                 // count=1 (user)
  g0[1] = lds;                                                // lds_addr
  g0[2] = (unsigned)(ga & 0xFFFFFFFFu);                       // global_addr lo
  g0[3] = (unsigned)((ga >> 32) & 0x01FFFFFFu) | (2u << 30);  // addr hi, type=2
  v8i g1;
  g1[0] = (int)((2u << 16) | (1u << 20) |                     // data_size, pad_en
                (pad_interval_code << 22) | (pad_amount_code << 25));
  g1[1] = (int)((tensor_d0 & 0xFFFFu) << 16);                 // tensor_dim0 lo16
  g1[2] = (int)((tensor_d0 >> 16) | ((tensor_d1 & 0xFFFFu) << 16));
  g1[3] = (int)((tensor_d1 >> 16) | (tile_d0 << 16));         // tile_dim0
  g1[4] = (int)(tile_d1 & 0xFFFFu);                           // tile_dim1, dim2=0
  g1[5] = (int)stride0;                                       // dim0 stride lo
  g1[6] = 0;
  g1[7] = 0;
  v4i g2 = {0, 0, 0, 0};
  v4i g3 = {0, 0, 0, 0};
  v8i g4 = {0, 0, 0, 0, 0, 0, 0, 0};
  __builtin_amdgcn_tensor_load_to_lds(g0, g1, g2, g3, g4, 0);
}

// ---------------------------------------------------------------------------
// WMMA f32 GEMM: C[40960,1024] = emb[40960,256] @ Wbig[256,1024]
// BM=64, BN=128, BK=16; 256 threads = 8 waves; wave (mw 0..3, nw 0..1)
// computes a 16x64 strip = 4 x (16x16) tiles via v_wmma_f32_16x16x4_f32.
// Tiles staged by TDM, double-buffered, TENSORcnt-paced.
// ---------------------------------------------------------------------------
#define LDA 20    // 16 dwords + 4 pad (TDM pad codes 3/3)
#define LDB 132   // 128 dwords + 4 pad (TDM pad codes 6/3)
__global__ __launch_bounds__(256) void gemm_kernel(
    const float* __restrict__ A, const float* __restrict__ Wb,
    float* __restrict__ Kd, float* __restrict__ Vd,
    float* __restrict__ L2d, float* __restrict__ E2d) {
  __shared__ __align__(16) float As[2 * 64 * LDA];
  __shared__ __align__(16) float Bs[2 * 16 * LDB];

  const int tid  = threadIdx.x;
  const int lane = tid & 31;
  const int wid  = tid >> 5;
  const int mw   = wid >> 1;            // 0..3
  const int nw   = wid & 1;             // 0..1
  const int l15  = lane & 15;
  const int kh   = (lane >> 4) << 1;    // lanes 0-15: K0/K1, lanes 16-31: K2/K3

  const int mBase = blockIdx.x * 64;
  const int nBlk  = blockIdx.y;         // 0..7
  float* outs[4] = {Kd, Vd, L2d, E2d};
  float* const outp = outs[nBlk >> 1];
  const int colGroupBase = (nBlk & 1) * 128;

  v8f acc[4];
  #pragma unroll
  for (int i = 0; i < 4; ++i) acc[i] = v8f{};

  // prologue: DMA chunk 0 into buffer 0 (wave 0 only; TDM is wave-level)
  if (wid == 0) {
    tdm_load_2d(A + (size_t)mBase * 256, lds_addr_of(&As[0]),
                256, 40960, 16, 64, 256, 3, 3);
    tdm_load_2d(Wb + (size_t)nBlk * 128, lds_addr_of(&Bs[0]),
                1024, 256, 128, 16, 1024, 6, 3);
  }

  for (int kc = 0; kc < 16; ++kc) {
    const int buf = kc & 1;
    if (wid == 0) {
      if (kc + 1 < 16) {
        const int kk = (kc + 1) * 16;
        // next chunk's target buffer was fully consumed before the previous
        // end-of-iteration barrier
        tdm_load_2d(A + (size_t)mBase * 256 + kk,
                    lds_addr_of(&As[(buf ^ 1) * 64 * LDA]),
                    256, 40960, 16, 64, 256, 3, 3);
        tdm_load_2d(Wb + (size_t)kk * 1024 + nBlk * 128,
                    lds_addr_of(&Bs[(buf ^ 1) * 16 * LDB]),
                    1024, 256, 128, 16, 1024, 6, 3);
        __builtin_amdgcn_s_wait_tensorcnt((short)2);  // chunk kc complete
      } else {
        __builtin_amdgcn_s_wait_tensorcnt((short)0);  // drain
      }
    }
    __syncthreads();   // LDS visibility of DMA'd tiles to all waves

    const float* As_c = As + buf * (64 * LDA);
    const float* Bs_c = Bs + buf * (16 * LDB);
    #pragma unroll
    for (int c = 0; c < 4; ++c) {
      const int k0 = c * 4;
      v2f a;
      a[0] = As_c[(mw * 16 + l15) * LDA + k0 + kh];
      a[1] = As_c[(mw * 16 + l15) * LDA + k0 + kh + 1];
      #pragma unroll
      for (int nt = 0; nt < 4; ++nt) {
        const int col = nw * 64 + nt * 16 + l15;
        v2f bb;
        bb[0] = Bs_c[(k0 + kh) * LDB + col];
        bb[1] = Bs_c[(k0 + kh + 1) * LDB + col];
        acc[nt] = __builtin_amdgcn_wmma_f32_16x16x4_f32(
            false, a, false, bb, (short)0, acc[nt], false, false);
      }
    }
    __syncthreads();   // all reads done before next iteration's DMA overwrites
  }

  // store: C/D layout: VGPR j -> row j (lanes 0-15) / row j+8 (lanes 16-31)
  const int rowBase = mBase + mw * 16 + ((lane >> 4) << 3);
  #pragma unroll
  for (int nt = 0; nt < 4; ++nt) {
    const int colL = colGroupBase + nw * 64 + nt * 16 + l15;
    #pragma unroll
    for (int j = 0; j < 8; ++j) {
      outp[(size_t)(rowBase + j) * 256 + colL] = acc[nt][j];
    }
  }
}

// ---------------------------------------------------------------------------
// Persistent per-batch decode kernel. grid=B (64), block=1024 (32 waves).
// ---------------------------------------------------------------------------
__global__ __launch_bounds__(1024) void step_kernel(
    const float* __restrict__ Kmat, const float* __restrict__ Vmat,
    const float* __restrict__ LK2, const float* __restrict__ E2,
    const float* __restrict__ fixed_ctx, const float* __restrict__ q0base,
    const int* __restrict__ n_steps_p, float* __restrict__ out_base) {
  const int b   = blockIdx.x;
  const int tid = threadIdx.x;
  const int T   = n_steps_p[0];

  float* const out_logp = out_base;                                // [B,T,G] f32
  int*   const out_pi   = (int*)(out_base + (size_t)B_ * T * G_);  // [B,T] i32 bits

  __shared__ __align__(16) float q_s[D_];
  __shared__ float num_s[H_ * G_];
  __shared__ __align__(16) float heads_s[D_];
  __shared__ float part_s[4 * D_];
  __shared__ float logits_s[G_];
  __shared__ float red_s[1024];
  __shared__ int   redi_s[1024];
  __shared__ float hmax_s[H_], hsum_s[H_];
  __shared__ unsigned char visited_s[G_];
  __shared__ int   prev_s, sel_s;
  __shared__ float gmax_s, lse_s;

  if (tid < G_) visited_s[tid] = 0;
  __syncthreads();

  for (int t = 0; t < T; ++t) {
    // ---- q = fixed_ctx + prev_embed @ W_step (precomputed rows) ----
    if (tid < D_) {
      if (t == 0) {
        q_s[tid] = q0base[(size_t)b * D_ + tid];
      } else {
        const int prev = prev_s;
        q_s[tid] = fixed_ctx[(size_t)b * D_ + tid] +
                   E2[((size_t)b * G_ + prev) * D_ + tid];
      }
    }
    __syncthreads();

    // ---- compat[h][g] = (q_h . K[b,g,h]) * inv_sqrt_dh, masked ----
    #pragma unroll
    for (int p = 0; p < 5; ++p) {
      const int idx = tid + p * 1024;      // 0..5119
      const int h = idx & 7, g = idx >> 3;
      const float4* kp = (const float4*)(Kmat + (((size_t)b * G_ + g) * D_ + h * DH_));
      const float4* qp = (const float4*)(q_s + h * DH_);
      float acc = 0.f;
      #pragma unroll
      for (int i = 0; i < 8; ++i) {
        const float4 kv = kp[i], qv = qp[i];
        acc += kv.x * qv.x + kv.y * qv.y + kv.z * qv.z + kv.w * qv.w;
      }
      num_s[h * G_ + g] = visited_s[g] ? NEG_INF_F : acc * INV_SQRT_DH;
    }
    __syncthreads();

    // ---- per-head softmax: max ----
    const int h = tid >> 7, j = tid & 127;   // 128 threads per head
    {
      float m = -3.4e38f;
      for (int g = j; g < G_; g += 128) m = fmaxf(m, num_s[h * G_ + g]);
      red_s[tid] = m;
      __syncthreads();
      for (int s = 64; s > 0; s >>= 1) {
        if (j < s) red_s[tid] = fmaxf(red_s[tid], red_s[tid + s]);
        __syncthreads();
      }
      if (j == 0) hmax_s[h] = red_s[tid];
      __syncthreads();
    }
    // ---- exp + sum (num_s becomes numerator) ----
    {
      const float mh = hmax_s[h];
      float ssum = 0.f;
      for (int g = j; g < G_; g += 128) {
        const float e = expf(num_s[h * G_ + g] - mh);
        num_s[h * G_ + g] = e;
        ssum += e;
      }
      red_s[tid] = ssum;
      __syncthreads();
      for (int s = 64; s > 0; s >>= 1) {
        if (j < s) red_s[tid] += red_s[tid + s];
        __syncthreads();
      }
      if (j == 0) hsum_s[h] = red_s[tid];
      __syncthreads();
    }

    // ---- heads[e] = sum_g num[h(e)][g] * V[b,g,e] / hsum ----
    {
      const int e = tid & 255, slice = tid >> 8;     // 4 g-slices of 160
      const float* np = num_s + (e >> 5) * G_;
      float acc = 0.f;
      const int g0 = slice * 160;
      for (int g = g0; g < g0 + 160; ++g)
        acc += np[g] * Vmat[((size_t)b * G_ + g) * D_ + e];
      part_s[slice * D_ + e] = acc;
      __syncthreads();
      if (tid < D_) {
        const float s4 = part_s[tid] + part_s[D_ + tid] +
                         part_s[2 * D_ + tid] + part_s[3 * D_ + tid];
        heads_s[tid] = s4 / hsum_s[tid >> 5];
      }
      __syncthreads();
    }

    // ---- logits[g] = 10*tanh(inv_sqrt_D * heads . LK2[b,g]), masked ----
    {
      const int k = tid & 7;                 // 8 lanes per g, 128 g per pass
      #pragma unroll
      for (int p = 0; p < 5; ++p) {
        const int gg = p * 128 + (tid >> 3);
        const float4* lp = (const float4*)(LK2 + (((size_t)b * G_ + gg) * D_ + k * DH_));
        const float4* hp = (const float4*)(heads_s + k * DH_);
        float acc = 0.f;
        #pragma unroll
        for (int i = 0; i < 8; ++i) {
          const float4 a = lp[i], c = hp[i];
          acc += a.x * c.x + a.y * c.y + a.z * c.z + a.w * c.w;
        }
        acc += __shfl_xor(acc, 1, 8);
        acc += __shfl_xor(acc, 2, 8);
        acc += __shfl_xor(acc, 4, 8);
        if (k == 0) {
          const float l = TANH_CLIP_F * tanhf(acc * INV_SQRT_D);
          logits_s[gg] = visited_s[gg] ? NEG_INF_F : l;
        }
      }
      __syncthreads();
    }

    // ---- argmax + max (first-index tie-break, matches jnp.argmax) ----
    {
      float v; int idx;
      if (tid < G_) { v = logits_s[tid]; idx = tid; }
      else          { v = -3.4e38f; idx = 0x7fffffff; }
      red_s[tid] = v; redi_s[tid] = idx;
      __syncthreads();
      for (int s = 512; s > 0; s >>= 1) {
        if (tid < s) {
          const float v2 = red_s[tid + s]; const int i2 = redi_s[tid + s];
          const float v1 = red_s[tid];     const int i1 = redi_s[tid];
          if (v2 > v1 || (v2 == v1 && i2 < i1)) { red_s[tid] = v2; redi_s[tid] = i2; }
        }
        __syncthreads();
      }
      if (tid == 0) { gmax_s = red_s[0]; sel_s = redi_s[0]; }
      __syncthreads();
    }
    // ---- log-sum-exp ----
    {
      red_s[tid] = (tid < G_) ? expf(logits_s[tid] - gmax_s) : 0.f;
      __syncthreads();
      for (int s = 512; s > 0; s >>= 1) {
        if (tid < s) red_s[tid] += red_s[tid + s];
        __syncthreads();
      }
      if (tid == 0) lse_s = logf(red_s[0]);
      __syncthreads();
    }

    // ---- emit, update state ----
    if (tid < G_)
      out_logp[((size_t)b * T + t) * G_ + tid] = logits_s[tid] - gmax_s - lse_s;
    if (tid == 0) {
      const int sel = sel_s;
      out_pi[(size_t)b * T + t] = sel;
      visited_s[sel] = 1;
      prev_s = sel;
    }
    __syncthreads();
  }
}

// ---------------------------------------------------------------------------
extern "C" void kernel_launch(void* const* d_in, const int* in_sizes, int n_in,
                              void* d_out, int out_size, void* d_ws, size_t ws_size,
                              hipStream_t stream) {
  (void)in_sizes; (void)n_in; (void)out_size; (void)ws_size;
  const float* emb     = (const float*)d_in[0];   // [64,640,256]
  const float* W_node  = (const float*)d_in[1];   // [256,768]
  const float* W_fixed = (const float*)d_in[2];   // [256,256]
  const float* W_step  = (const float*)d_in[3];   // [256,256]
  const float* W_out   = (const float*)d_in[4];   // [256,256]
  const int*   n_steps = (const int*)d_in[5];     // scalar

  float* ws = (float*)d_ws;
  const size_t BGD = (size_t)B_ * G_ * D_;        // 10,485,760
  float* Km   = ws;
  float* Vm   = ws + BGD;
  float* L2m  = ws + 2 * BGD;
  float* E2m  = ws + 3 * BGD;
  float* Wbig = ws + 4 * BGD;                     // 256*1024
  float* fctx = Wbig + 256 * 1024;                // 64*256
  float* q0b  = fctx + B_ * D_;                   // 64*256

  wprep_kernel<<<dim3(256), dim3(256), 0, stream>>>(W_node, W_step, W_out, Wbig);
  ctx_kernel<<<dim3(B_), dim3(256), 0, stream>>>(emb, W_fixed, W_step, fctx, q0b);
  gemm_kernel<<<dim3(640, 8), dim3(256), 0, stream>>>(emb, Wbig, Km, Vm, L2m, E2m);
  step_kernel<<<dim3(B_), dim3(1024), 0, stream>>>(Km, Vm, L2m, E2m, fctx, q0b,
                                                   n_steps, (float*)d_out);
}